// GAE_1898375544939
// MI455X (gfx1250) — compile-verified
//
#include <hip/hip_runtime.h>
#include <hip/hip_bf16.h>

typedef __attribute__((ext_vector_type(2))) float v2f;
typedef __attribute__((ext_vector_type(8))) float v8f;

// D = A(16x4) * B(4x16) + C, f32 WMMA (exact f32 math, matches reference)
static __device__ __forceinline__ v8f wmma4(v2f a, v2f b, v8f c) {
  return __builtin_amdgcn_wmma_f32_16x16x4_f32(
      /*neg_a=*/false, a, /*neg_b=*/false, b,
      /*c_mod=*/(short)0, c, /*reuse_a=*/false, /*reuse_b=*/false);
}

static __device__ __forceinline__ v8f v8f_zero() {
  return v8f{0.f, 0.f, 0.f, 0.f, 0.f, 0.f, 0.f, 0.f};
}

// load two consecutive floats as one b64
static __device__ __forceinline__ v2f ldpair(const float* p) {
  return *reinterpret_cast<const v2f*>(p);
}

// ---------------- elementwise / graph kernels ----------------

__global__ void k_zero(float* __restrict__ p, int n) {
  int i = blockIdx.x * blockDim.x + threadIdx.x;
  if (i < n) p[i] = 0.0f;
}

__global__ void k_degree(const int* __restrict__ src, const int* __restrict__ dst,
                         float* __restrict__ dout, float* __restrict__ din, int E) {
  int i = blockIdx.x * blockDim.x + threadIdx.x;
  if (i < E) {
    unsafeAtomicAdd(&dout[src[i]], 1.0f);
    unsafeAtomicAdd(&din[dst[i]], 1.0f);
  }
}

__global__ void k_norm(float* __restrict__ p, int n) {
  int i = blockIdx.x * blockDim.x + threadIdx.x;
  if (i < n) {
    float d = fmaxf(p[i], 1.0f);
    p[i] = 1.0f / sqrtf(d);
  }
}

template <int F, int LOGF>
__global__ void k_scatter(const int* __restrict__ src, const int* __restrict__ dst,
                          const float* __restrict__ h, float* __restrict__ agg, int E) {
  int idx = blockIdx.x * blockDim.x + threadIdx.x;
  if (idx < E * F) {
    int e = idx >> LOGF;
    int f = idx & (F - 1);
    unsafeAtomicAdd(&agg[(dst[e] << LOGF) + f], h[(src[e] << LOGF) + f]);
  }
}

template <int F, int LOGF>
__global__ void k_bias_relu(float* __restrict__ h, const float* __restrict__ ndst,
                            const float* __restrict__ bias, int n) {
  int idx = blockIdx.x * blockDim.x + threadIdx.x;
  if (idx < n) {
    int i = idx >> LOGF;
    int f = idx & (F - 1);
    h[idx] = fmaxf(h[idx] * ndst[i] + bias[f], 0.0f);
  }
}

__global__ void k_zmask(float* __restrict__ zb, const float* __restrict__ ndst,
                        const float* __restrict__ bias, const float* __restrict__ mask, int n) {
  int idx = blockIdx.x * blockDim.x + threadIdx.x;
  if (idx < n) {
    int i = idx >> 6;
    int f = idx & 63;
    zb[idx] = (zb[idx] * ndst[i] + bias[f]) * mask[idx];
  }
}

// ---------------- WMMA GEMM: out[M,Nc] = (A .* scale[:,None]) @ W ----------------
// One wave (32 threads) computes a 32x64 block: 2 M-tiles x 4 N-tiles.
// K, Nc compile-time: static trip count, no spill-prone dynamic loop.
template <int K, int Nc>
__global__ void k_gemm_rowscale(const float* __restrict__ A, const float* __restrict__ scale,
                                const float* __restrict__ W, float* __restrict__ out) {
  const int lane = threadIdx.x & 31;
  const int half = lane >> 4;   // 0: K=k0,k0+1 / M=0..7 ; 1: K=k0+2,k0+3 / M=8..15
  const int l16  = lane & 15;
  const int m0 = blockIdx.y << 5;
  const int n0 = blockIdx.x << 6;

  const int r0 = m0 + l16;
  const int r1 = m0 + 16 + l16;
  const float s0 = scale[r0];
  const float s1 = scale[r1];
  const float* Ar0 = A + r0 * K;
  const float* Ar1 = A + r1 * K;

  v8f acc[2][4];
#pragma unroll
  for (int i = 0; i < 2; ++i)
#pragma unroll
    for (int j = 0; j < 4; ++j) acc[i][j] = v8f_zero();

#pragma unroll 4
  for (int k0 = 0; k0 < K; k0 += 4) {
    const int kb = k0 + (half << 1);
    v2f a0 = ldpair(Ar0 + kb);   // one b64 load per fragment
    v2f a1 = ldpair(Ar1 + kb);
    a0.x *= s0; a0.y *= s0;
    a1.x *= s1; a1.y *= s1;
#pragma unroll
    for (int j = 0; j < 4; ++j) {
      const int col = n0 + (j << 4) + l16;
      v2f b;
      b.x = W[kb * Nc + col];          // Nc-strided: stays scalar
      b.y = W[(kb + 1) * Nc + col];
      acc[0][j] = wmma4(a0, b, acc[0][j]);
      acc[1][j] = wmma4(a1, b, acc[1][j]);
    }
  }

#pragma unroll
  for (int i = 0; i < 2; ++i) {
    const int mb = m0 + (i << 4) + (half << 3);
#pragma unroll
    for (int j = 0; j < 4; ++j) {
      const int col = n0 + (j << 4) + l16;
#pragma unroll
      for (int r = 0; r < 8; ++r)
        out[(mb + r) * Nc + col] = acc[i][j][r];
    }
  }
}

// ---------------- decode: out = z @ z^T, z:[N,64], NT stores for 1GiB stream ----------------
__global__ void k_zzt(const float* __restrict__ z, float* __restrict__ out, int N) {
  const int lane = threadIdx.x & 31;
  const int half = lane >> 4;
  const int l16  = lane & 15;
  const int m0 = blockIdx.y << 5;   // 32 rows
  const int n0 = blockIdx.x << 6;   // 64 cols

  const float* zr0 = z + (m0 + l16) * 64;
  const float* zr1 = z + (m0 + 16 + l16) * 64;
  const float* zc0 = z + (n0 + l16) * 64;       // B fragments: rows n of z
  const float* zc1 = zc0 + 16 * 64;
  const float* zc2 = zc0 + 32 * 64;
  const float* zc3 = zc0 + 48 * 64;

  v8f acc[2][4];
#pragma unroll
  for (int i = 0; i < 2; ++i)
#pragma unroll
    for (int j = 0; j < 4; ++j) acc[i][j] = v8f_zero();

#pragma unroll
  for (int k0 = 0; k0 < 64; k0 += 4) {
    const int kb = k0 + (half << 1);
    v2f a0 = ldpair(zr0 + kb);   // all fragment pairs contiguous -> b64 loads
    v2f a1 = ldpair(zr1 + kb);
    v2f b0 = ldpair(zc0 + kb);
    v2f b1 = ldpair(zc1 + kb);
    v2f b2 = ldpair(zc2 + kb);
    v2f b3 = ldpair(zc3 + kb);
    acc[0][0] = wmma4(a0, b0, acc[0][0]);
    acc[1][0] = wmma4(a1, b0, acc[1][0]);
    acc[0][1] = wmma4(a0, b1, acc[0][1]);
    acc[1][1] = wmma4(a1, b1, acc[1][1]);
    acc[0][2] = wmma4(a0, b2, acc[0][2]);
    acc[1][2] = wmma4(a1, b2, acc[1][2]);
    acc[0][3] = wmma4(a0, b3, acc[0][3]);
    acc[1][3] = wmma4(a1, b3, acc[1][3]);
  }

#pragma unroll
  for (int i = 0; i < 2; ++i) {
    const int mb = m0 + (i << 4) + (half << 3);
#pragma unroll
    for (int j = 0; j < 4; ++j) {
      const int col = n0 + (j << 4) + l16;
#pragma unroll
      for (int r = 0; r < 8; ++r)
        __builtin_nontemporal_store(acc[i][j][r], &out[(size_t)(mb + r) * N + col]);
    }
  }
}

// ---------------- launch ----------------

extern "C" void kernel_launch(void* const* d_in, const int* in_sizes, int n_in,
                              void* d_out, int out_size, void* d_ws, size_t ws_size,
                              hipStream_t stream) {
  const int N = 16384, E = 524288;
  const float* x    = (const float*)d_in[0];
  const int*   src  = (const int*)d_in[1];
  const int*   dst  = (const int*)d_in[2];
  const float* W1   = (const float*)d_in[3];
  const float* b1   = (const float*)d_in[4];
  const float* W2   = (const float*)d_in[5];
  const float* b2   = (const float*)d_in[6];
  const float* mask = (const float*)d_in[7];
  float* out = (float*)d_out;

  // small scratch in d_ws (~8.5 MB)
  float* wsf   = (float*)d_ws;
  float* nsrc  = wsf;                 // N   deg_out -> norm_src
  float* ndst  = wsf + N;             // N   deg_in  -> norm_dst
  float* hlin2 = wsf + 2 * N;         // 64N
  float* zbuf  = wsf + 2 * N + 64 * N;// 64N agg2 -> z

  // big scratch in front of the 1 GiB output (decode overwrites it last)
  float* hlin1 = out;                 // 256N
  float* agg1  = out + 256 * N;       // 256N, becomes h1 in place

  auto cdiv = [](int a, int b) { return (a + b - 1) / b; };

  // zero accumulators (deterministic per call)
  k_zero<<<cdiv(2 * N, 256), 256, 0, stream>>>(nsrc, 2 * N);
  k_zero<<<cdiv(256 * N, 256), 256, 0, stream>>>(agg1, 256 * N);
  k_zero<<<cdiv(64 * N, 256), 256, 0, stream>>>(zbuf, 64 * N);

  // degrees -> norms
  k_degree<<<cdiv(E, 256), 256, 0, stream>>>(src, dst, nsrc, ndst, E);
  k_norm<<<cdiv(2 * N, 256), 256, 0, stream>>>(nsrc, 2 * N);

  // layer 1: (x .* nsrc) @ W1 -> scatter -> relu(.*ndst + b1)
  k_gemm_rowscale<512, 256><<<dim3(256 / 64, N / 32), 32, 0, stream>>>(x, nsrc, W1, hlin1);
  k_scatter<256, 8><<<E * 256 / 256, 256, 0, stream>>>(src, dst, hlin1, agg1, E);
  k_bias_relu<256, 8><<<N * 256 / 256, 256, 0, stream>>>(agg1, ndst, b1, N * 256);

  // layer 2: (h1 .* nsrc) @ W2 -> scatter -> (.*ndst + b2) * drop_mask
  k_gemm_rowscale<256, 64><<<dim3(64 / 64, N / 32), 32, 0, stream>>>(agg1, nsrc, W2, hlin2);
  k_scatter<64, 6><<<E * 64 / 256, 256, 0, stream>>>(src, dst, hlin2, zbuf, E);
  k_zmask<<<N * 64 / 256, 256, 0, stream>>>(zbuf, ndst, b2, mask, N * 64);

  // decode: out = z @ z^T  (store-bandwidth bound; WMMA f32)
  k_zzt<<<dim3(N / 64, N / 32), 32, 0, stream>>>(zbuf, out, N);
}